// AttentionConstrainedLoss_54855322304566
// MI455X (gfx1250) — compile-verified
//
#include <hip/hip_runtime.h>
#include <stdint.h>

typedef __attribute__((ext_vector_type(2))) float v2f;
typedef __attribute__((ext_vector_type(8))) float v8f;
typedef __attribute__((ext_vector_type(4))) int v4i;

#define HH 40
#define WW 40
#define G_CELLS 1600
#define B_SZ 16
#define M_BOX 64
#define D_DIM 2048

#if defined(__has_builtin)
#if __has_builtin(__builtin_amdgcn_global_load_async_to_lds_b128) && \
    __has_builtin(__builtin_amdgcn_s_wait_asynccnt)
#define USE_ASYNC_LDS 1
#endif
#endif
#ifndef USE_ASYNC_LDS
#define USE_ASYNC_LDS 0
#endif

// ---------------------------------------------------------------------------
// K1: per-batch sequential box->grid assignment (flags), registers + LDS argmin
// ---------------------------------------------------------------------------
__global__ __launch_bounds__(320) void k_flags(const float* __restrict__ gt,
                                               int* __restrict__ flags_ws) {
  const int b = blockIdx.x;
  const int t = threadIdx.x;
  __shared__ unsigned long long keys[512];
  __shared__ int anyViol;

  float gx[5], gy[5];
  int flag[5];
#pragma unroll
  for (int j = 0; j < 5; ++j) {
    int g = t + 320 * j;
    int wix = g % WW, hix = g / WW;
    gx[j] = ((float)wix + 0.5f) / 40.0f * 102.4f + (-51.2f);
    gy[j] = ((float)hix + 0.5f) / 40.0f * 102.4f + (-51.2f);
    flag[j] = -1;
  }

  for (int m = 0; m < M_BOX; ++m) {
    const float* bb = gt + ((size_t)b * M_BOX + m) * 7;
    float cx = bb[0], cy = bb[1];
    float bl = bb[3], bw = bb[4];
    float ang = bb[6];
    float rl = fminf(fmaxf(2.56f / bl, 1.0f), 6.0f);
    float rw = fminf(fmaxf(2.56f / bw, 1.0f), 6.0f);
    float ex = bl * rl, ey = bw * rw;  // effective dims (x-extent, y-extent)
    float cc = cosf(ang), ss = sinf(ang);

    const float cnx[4] = {-0.5f, -0.5f, 0.5f, 0.5f};
    const float cny[4] = {-0.5f, 0.5f, 0.5f, -0.5f};
    float crx[4], cry[4];
#pragma unroll
    for (int k = 0; k < 4; ++k) {
      float px = cnx[k] * ex, py = cny[k] * ey;
      crx[k] = cc * px - ss * py + cx;
      cry[k] = ss * px + cc * py + cy;
    }
    float edx[4], edy[4];
#pragma unroll
    for (int k = 0; k < 4; ++k) {
      edx[k] = crx[(k + 1) & 3] - crx[k];
      edy[k] = cry[(k + 1) & 3] - cry[k];
    }

    unsigned inside_bits = 0;
    unsigned long long lkey = ~0ULL;
#pragma unroll
    for (int j = 0; j < 5; ++j) {
      int g = t + 320 * j;
      bool allp = true, alln = true;
#pragma unroll
      for (int k = 0; k < 4; ++k) {
        float rx = gx[j] - crx[k], ry = gy[j] - cry[k];
        float cr = edx[k] * ry - edy[k] * rx;
        allp = allp && (cr >= 0.0f);
        alln = alln && (cr <= 0.0f);
      }
      if (allp || alln) inside_bits |= (1u << j);
      float dx = gx[j] - cx, dy = gy[j] - cy;
      float dist = dx * dx + dy * dy;
      unsigned long long key =
          ((unsigned long long)__float_as_uint(dist) << 32) | (unsigned)g;
      lkey = (key < lkey) ? key : lkey;
    }

    keys[t] = lkey;
    if (t < 192) keys[320 + t] = ~0ULL;
    if (t == 0) anyViol = 0;
    __syncthreads();
    for (int sft = 256; sft > 0; sft >>= 1) {
      if (t < sft) {
        unsigned long long o = keys[t + sft];
        if (o < keys[t]) keys[t] = o;
      }
      __syncthreads();
    }
    int argg = (int)(keys[0] & 0xFFFFFFFFu);

    int viol = 0;
    unsigned maskb = 0;
#pragma unroll
    for (int j = 0; j < 5; ++j) {
      int g = t + 320 * j;
      bool msk = (((inside_bits >> j) & 1u) != 0u) || (g == argg);
      if (msk) {
        maskb |= (1u << j);
        if (flag[j] != -1) viol = 1;
      }
    }
    if (viol) atomicOr(&anyViol, 1);
    __syncthreads();
    bool all_free = (anyViol == 0);
#pragma unroll
    for (int j = 0; j < 5; ++j) {
      if ((maskb >> j) & 1u) flag[j] = (all_free || flag[j] == -1) ? m : -1;
    }
    __syncthreads();  // protect keys/anyViol for next iteration
  }

#pragma unroll
  for (int j = 0; j < 5; ++j) flags_ws[b * G_CELLS + t + 320 * j] = flag[j];
}

// ---------------------------------------------------------------------------
// K2: streaming variance over D=2048 (memory-bound pass, 1 wave per row).
// CDNA5 path: stage each 8KB row through LDS with global_load_async_to_lds_b128
// (4 chunks x 4 per-lane b128 ops), drained in-order with s_wait_asynccnt.
// ---------------------------------------------------------------------------
#if USE_ASYNC_LDS
typedef __attribute__((address_space(1))) void* as1_vp;
typedef __attribute__((address_space(3))) void* as3_vp;
typedef __attribute__((address_space(1))) v4i* as1_v4i;
typedef __attribute__((address_space(3))) v4i* as3_v4i;

__global__ __launch_bounds__(256) void k_var(const float* __restrict__ atten,
                                             float* __restrict__ v_ws) {
  __shared__ float buf[8][4][512];  // 8 waves x 4 chunks x 2KB = 64KB staging
  const int lane = threadIdx.x & 31;
  const int wav = threadIdx.x >> 5;
  const int row = blockIdx.x * 8 + wav;
  const float* base = atten + (size_t)row * D_DIM;

  // Issue the whole row: 16 async b128 copies per lane (ASYNCcnt = 16).
#pragma unroll
  for (int c = 0; c < 4; ++c) {
#pragma unroll
    for (int q = 0; q < 4; ++q) {
      const float* g = base + c * 512 + q * 128 + lane * 4;
      float* l = &buf[wav][c][q * 128 + lane * 4];
      __builtin_amdgcn_global_load_async_to_lds_b128(
          (as1_v4i)(as1_vp)(void*)g, (as3_v4i)(as3_vp)(void*)l, 0, 0);
    }
  }

  float s = 0.0f, s2 = 0.0f;
#define CONSUME_CHUNK(C)                                      \
  do {                                                        \
    asm volatile("" ::: "memory");                            \
    const float* l = &buf[wav][C][0];                         \
    _Pragma("unroll") for (int q = 0; q < 4; ++q) {           \
      float4 x = *(const float4*)(l + q * 128 + lane * 4);    \
      s += x.x + x.y + x.z + x.w;                             \
      s2 += x.x * x.x + x.y * x.y + x.z * x.z + x.w * x.w;    \
    }                                                         \
  } while (0)

  // Async completions are in-order: cnt<=12 releases chunk0, <=8 chunk1, ...
  __builtin_amdgcn_s_wait_asynccnt(12);
  CONSUME_CHUNK(0);
  __builtin_amdgcn_s_wait_asynccnt(8);
  CONSUME_CHUNK(1);
  __builtin_amdgcn_s_wait_asynccnt(4);
  CONSUME_CHUNK(2);
  __builtin_amdgcn_s_wait_asynccnt(0);
  CONSUME_CHUNK(3);
#undef CONSUME_CHUNK

#pragma unroll
  for (int off = 16; off > 0; off >>= 1) {
    s += __shfl_xor(s, off, 32);
    s2 += __shfl_xor(s2, off, 32);
  }
  if (lane == 0) {
    float var =
        (s2 - s * s * (1.0f / (float)D_DIM)) * (1.0f / (float)(D_DIM - 1));
    v_ws[row] = var;
  }
}
#else
__global__ __launch_bounds__(256) void k_var(const float* __restrict__ atten,
                                             float* __restrict__ v_ws) {
  const int lane = threadIdx.x & 31;
  const int wav = threadIdx.x >> 5;
  const int row = blockIdx.x * 8 + wav;
  const float* base = atten + (size_t)row * D_DIM;

  float s = 0.0f, s2 = 0.0f;
#pragma unroll
  for (int i = 0; i < 16; ++i) {
    const float4 x = *(const float4*)(base + i * 128 + lane * 4);
    if (i + 4 < 16)
      __builtin_prefetch((const void*)(base + (i + 4) * 128 + lane * 4), 0, 0);
    s += x.x + x.y + x.z + x.w;
    s2 += x.x * x.x + x.y * x.y + x.z * x.z + x.w * x.w;
  }
#pragma unroll
  for (int off = 16; off > 0; off >>= 1) {
    s += __shfl_xor(s, off, 32);
    s2 += __shfl_xor(s2, off, 32);
  }
  if (lane == 0) {
    float var =
        (s2 - s * s * (1.0f / (float)D_DIM)) * (1.0f / (float)(D_DIM - 1));
    v_ws[row] = var;
  }
}
#endif

// ---------------------------------------------------------------------------
// K3: per-batch grouped sums via V_WMMA_F32_16X16X4_F32 (exact f32) + counts
//     A = onehot tile (16 m x 4 g, entries 0/1), B[k][n] = v[g0+k] broadcast
// ---------------------------------------------------------------------------
__global__ __launch_bounds__(128) void k_sums(const int* __restrict__ flags_ws,
                                              const float* __restrict__ v_ws,
                                              float* __restrict__ sums_ws,
                                              int* __restrict__ counts_ws) {
  const int b = blockIdx.x;
  const int t = threadIdx.x;
  __shared__ int cnt[M_BOX];
  if (t < M_BOX) cnt[t] = 0;
  __syncthreads();

  const int* fl = flags_ws + b * G_CELLS;
  const float* vv = v_ws + b * G_CELLS;

  for (int g = t; g < G_CELLS; g += 128) {
    int f = fl[g];
    if (f >= 0) atomicAdd(&cnt[f], 1);
  }

  const int lane = t & 31;
  const int wav = t >> 5;      // m-tile index (4 tiles of 16)
  const int mloc = lane & 15;  // M within tile for A
  const int m = wav * 16 + mloc;
  const int hi = lane >> 4;  // K-pair select: lanes 0-15 -> K=0,1 ; 16-31 -> K=2,3

  v8f c = {};
  for (int g0 = 0; g0 < G_CELLS; g0 += 4) {
    int g = g0 + 2 * hi;
    int f0 = fl[g], f1 = fl[g + 1];
    v2f a, bv;
    a.x = (f0 == m) ? 1.0f : 0.0f;
    a.y = (f1 == m) ? 1.0f : 0.0f;
    bv.x = vv[g];
    bv.y = vv[g + 1];
    c = __builtin_amdgcn_wmma_f32_16x16x4_f32(false, a, false, bv, (short)0, c,
                                              false, false);
  }
  // D layout: vgpr r, lanes 0-15 -> M=r (N=lane), lanes 16-31 -> M=r+8.
  // Every column N holds the same value; lanes with N==0 write out.
  if (mloc == 0) {
#pragma unroll
    for (int r = 0; r < 8; ++r)
      sums_ws[b * M_BOX + wav * 16 + hi * 8 + r] = c[r];
  }
  __syncthreads();
  if (t < M_BOX) counts_ws[b * M_BOX + t] = cnt[t];
}

// ---------------------------------------------------------------------------
// K4: final scalar reduction
// ---------------------------------------------------------------------------
__global__ __launch_bounds__(256) void k_final(const float* __restrict__ sums_ws,
                                               const int* __restrict__ counts_ws,
                                               float* __restrict__ out) {
  __shared__ float redl[256];
  __shared__ float redp[256];
  const int t = threadIdx.x;
  float loss = 0.0f, pos = 0.0f;
  for (int i = t; i < B_SZ * M_BOX; i += 256) {
    int cct = counts_ws[i];
    if (cct > 0) {
      loss -= sums_ws[i] / (float)cct;
      pos += 1.0f;
    }
  }
  redl[t] = loss;
  redp[t] = pos;
  __syncthreads();
  for (int s = 128; s > 0; s >>= 1) {
    if (t < s) {
      redl[t] += redl[t + s];
      redp[t] += redp[t + s];
    }
    __syncthreads();
  }
  if (t == 0) out[0] = redl[0] / fmaxf(redp[0], 1.0f);
}

// ---------------------------------------------------------------------------
extern "C" void kernel_launch(void* const* d_in, const int* in_sizes, int n_in,
                              void* d_out, int out_size, void* d_ws,
                              size_t ws_size, hipStream_t stream) {
  const float* atten = (const float*)d_in[0];  // (16,1600,2048) f32
  const float* gt = (const float*)d_in[1];     // (16,64,7) f32
  float* out = (float*)d_out;                  // scalar f32

  char* ws = (char*)d_ws;
  int* flags_ws = (int*)(ws + 0);       // 16*1600 ints
  float* v_ws = (float*)(ws + 102400);  // 16*1600 f32
  float* sums_ws = (float*)(ws + 204800);  // 16*64 f32
  int* counts_ws = (int*)(ws + 208896);    // 16*64 ints

  k_flags<<<B_SZ, 320, 0, stream>>>(gt, flags_ws);
  k_var<<<(B_SZ * G_CELLS) / 8, 256, 0, stream>>>(atten, v_ws);
  k_sums<<<B_SZ, 128, 0, stream>>>(flags_ws, v_ws, sums_ws, counts_ws);
  k_final<<<1, 256, 0, stream>>>(sums_ws, counts_ws, out);
}